// GINLayer_64862596104930
// MI455X (gfx1250) — compile-verified
//
#include <hip/hip_runtime.h>
#include <hip/hip_bf16.h>

#define NN 100000
#define NE 1600000
#define DD 64
#define BN_EPS 1e-5f
#define NTILES (NN / 16)          // 6250, exact
#define LDSP 68                   // padded row stride (floats) -> conflict-free column reads

typedef __attribute__((ext_vector_type(2))) float v2f;
typedef __attribute__((ext_vector_type(8))) float v8f;

__device__ __forceinline__ void gatomic_add(float* p, float v) {
    // lowers to global_atomic_add_u32/f32 no-return (result unused)
    __hip_atomic_fetch_add(p, v, __ATOMIC_RELAXED, __HIP_MEMORY_SCOPE_AGENT);
}

// ---------------- kernel 0: h = (1+eps)*features ; zero BN stat accumulators ----------------
__global__ void __launch_bounds__(256) k_init(const float* __restrict__ feat,
                                              const float* __restrict__ eps,
                                              float* __restrict__ h,
                                              float* __restrict__ stats) {
    int i = blockIdx.x * 256 + threadIdx.x;
    if (blockIdx.x == 0 && threadIdx.x < 128) stats[threadIdx.x] = 0.0f;
    float e = 1.0f + eps[0];
    int idx = i * 4;
    float4 f = *(const float4*)(feat + idx);
    f.x *= e; f.y *= e; f.z *= e; f.w *= e;
    *(float4*)(h + idx) = f;
}

// ---------------- kernel 1: scatter-sum  h[dst] += feat[src] ----------------
__global__ void __launch_bounds__(256) k_scatter(const float* __restrict__ feat,
                                                 const int* __restrict__ src,
                                                 const int* __restrict__ dst,
                                                 float* __restrict__ h) {
    int i = blockIdx.x * 256 + threadIdx.x;       // 0 .. NE*64-1  (fits int32)
    int e = i >> 6;
    int d = i & 63;
    int s = src[e];
    int t = dst[e];
    gatomic_add(&h[t * DD + d], feat[s * DD + d]);
}

// ---------------- kernel 2: fused MLP per 16-row tile, in-place, f32 WMMA ----------------
// A-fragment (16x4 f32): lane m=lane&15 holds row m; vgpr j holds K = k0 + 2*(lane>>4) + j
// B-fragment (4x16 f32): lane n=lane&15 holds col n; vgpr j holds K = k0 + 2*(lane>>4) + j
// C/D (16x16 f32):       acc[v] = D[v + 8*(lane>>4)][lane&15]
__global__ void __launch_bounds__(128) k_mlp(float* __restrict__ h,
                                             const float* __restrict__ W1,
                                             const float* __restrict__ b1,
                                             const float* __restrict__ W2,
                                             const float* __restrict__ b2) {
    __shared__ float tile[4][16 * LDSP];
    int wave = threadIdx.x >> 5;
    int lane = threadIdx.x & 31;
    int t = blockIdx.x * 4 + wave;
    if (t >= NTILES) return;                       // uniform per wave -> EXEC all-1 for WMMA
    float* A = &tile[wave][0];
    int row0 = t * 16;

    // load 16x64 tile -> LDS (coalesced float4)
    for (int i = lane; i < 256; i += 32) {
        int r = i >> 4, c = i & 15;
        *(float4*)(A + r * LDSP + c * 4) = *(const float4*)(h + (row0 + r) * DD + c * 4);
    }
    asm volatile("s_wait_dscnt 0" ::: "memory");

    int m  = lane & 15;
    int kh = (lane >> 4) << 1;
    int rbase = (lane >> 4) * 8;

    // ---- GEMM1: ReLU(h @ W1 + b1) ----
    v8f acc[4];
#pragma unroll
    for (int nt = 0; nt < 4; ++nt) {
        float bias = b1[nt * 16 + m];              // column bias: same for all 8 C rows
        acc[nt] = (v8f){bias, bias, bias, bias, bias, bias, bias, bias};
    }
    for (int k0 = 0; k0 < DD; k0 += 4) {
        v2f a = *(const v2f*)(A + m * LDSP + k0 + kh);
#pragma unroll
        for (int nt = 0; nt < 4; ++nt) {
            int n = nt * 16 + m;
            v2f b;
            b.x = W1[(k0 + kh) * DD + n];
            b.y = W1[(k0 + kh + 1) * DD + n];
            acc[nt] = __builtin_amdgcn_wmma_f32_16x16x4_f32(
                false, a, false, b, (short)0, acc[nt], false, false);
        }
    }
    // ReLU -> LDS (h1 becomes next A operand)
#pragma unroll
    for (int nt = 0; nt < 4; ++nt)
#pragma unroll
        for (int v = 0; v < 8; ++v)
            A[(rbase + v) * LDSP + nt * 16 + m] = fmaxf(acc[nt][v], 0.0f);
    asm volatile("s_wait_dscnt 0" ::: "memory");

    // ---- GEMM2: h1 @ W2 + b2 ----
#pragma unroll
    for (int nt = 0; nt < 4; ++nt) {
        float bias = b2[nt * 16 + m];
        acc[nt] = (v8f){bias, bias, bias, bias, bias, bias, bias, bias};
    }
    for (int k0 = 0; k0 < DD; k0 += 4) {
        v2f a = *(const v2f*)(A + m * LDSP + k0 + kh);
#pragma unroll
        for (int nt = 0; nt < 4; ++nt) {
            int n = nt * 16 + m;
            v2f b;
            b.x = W2[(k0 + kh) * DD + n];
            b.y = W2[(k0 + kh + 1) * DD + n];
            acc[nt] = __builtin_amdgcn_wmma_f32_16x16x4_f32(
                false, a, false, b, (short)0, acc[nt], false, false);
        }
    }
    // stage result in LDS, then coalesced in-place write-back
#pragma unroll
    for (int nt = 0; nt < 4; ++nt)
#pragma unroll
        for (int v = 0; v < 8; ++v)
            A[(rbase + v) * LDSP + nt * 16 + m] = acc[nt][v];
    asm volatile("s_wait_dscnt 0" ::: "memory");
    for (int i = lane; i < 256; i += 32) {
        int r = i >> 4, c = i & 15;
        *(float4*)(h + (row0 + r) * DD + c * 4) = *(const float4*)(A + r * LDSP + c * 4);
    }
}

// ---------------- kernel 3: BN column sums (sum, sumsq) ----------------
#define STAT_BLOCKS 512
#define ROWS_PB 196               // 512*196 >= 100000
__global__ void __launch_bounds__(256) k_stats(const float* __restrict__ h,
                                               float* __restrict__ stats) {
    __shared__ float ssum[64], ssq[64];
    int tid = threadIdx.x;
    if (tid < 64) { ssum[tid] = 0.0f; ssq[tid] = 0.0f; }
    __syncthreads();
    int col = tid & 63;
    int rend = (blockIdx.x + 1) * ROWS_PB;
    if (rend > NN) rend = NN;
    float s = 0.0f, q = 0.0f;
    for (int r = blockIdx.x * ROWS_PB + (tid >> 6); r < rend; r += 4) {
        float x = h[r * DD + col];
        s += x; q += x * x;
    }
    atomicAdd(&ssum[col], s);      // ds_add_f32
    atomicAdd(&ssq[col], q);
    __syncthreads();
    if (tid < 64) {
        gatomic_add(&stats[tid], ssum[tid]);
        gatomic_add(&stats[64 + tid], ssq[tid]);
    }
}

// ---------------- kernel 4: BN apply + ReLU + residual ----------------
__global__ void __launch_bounds__(256) k_final(const float* __restrict__ feat,
                                               const float* __restrict__ h,
                                               const float* __restrict__ gamma,
                                               const float* __restrict__ beta,
                                               const float* __restrict__ stats,
                                               float* __restrict__ out) {
    int i = blockIdx.x * 256 + threadIdx.x;
    int idx = i * 4;
    const float invN = 1.0f / (float)NN;
    float4 hv = *(const float4*)(h + idx);
    float4 fv = *(const float4*)(feat + idx);
    float hin[4] = {hv.x, hv.y, hv.z, hv.w};
    float fin[4] = {fv.x, fv.y, fv.z, fv.w};
    float o[4];
    int d0 = idx & 63;             // multiple of 4, so d0+j stays within the row
#pragma unroll
    for (int j = 0; j < 4; ++j) {
        int d = d0 + j;
        float mean = stats[d] * invN;
        float var  = stats[64 + d] * invN - mean * mean;
        float xs = (hin[j] - mean) * rsqrtf(var + BN_EPS) * gamma[d] + beta[d];
        o[j] = fin[j] + fmaxf(xs, 0.0f);
    }
    float4 ov = {o[0], o[1], o[2], o[3]};
    *(float4*)(out + idx) = ov;
}

extern "C" void kernel_launch(void* const* d_in, const int* in_sizes, int n_in,
                              void* d_out, int out_size, void* d_ws, size_t ws_size,
                              hipStream_t stream) {
    const float* feat  = (const float*)d_in[0];
    const int*   src   = (const int*)d_in[1];
    const int*   dst   = (const int*)d_in[2];
    const float* eps   = (const float*)d_in[3];
    const float* W1    = (const float*)d_in[4];
    const float* b1    = (const float*)d_in[5];
    const float* W2    = (const float*)d_in[6];
    const float* b2    = (const float*)d_in[7];
    const float* gamma = (const float*)d_in[8];
    const float* beta  = (const float*)d_in[9];
    float* out = (float*)d_out;

    float* h     = (float*)d_ws;            // NN*DD floats (25.6 MB)
    float* stats = h + (size_t)NN * DD;     // 128 floats: [0..63]=sum, [64..127]=sumsq

    k_init   <<<(NN * DD / 4) / 256, 256, 0, stream>>>(feat, eps, h, stats);
    k_scatter<<<(NE * DD) / 256,     256, 0, stream>>>(feat, src, dst, h);
    k_mlp    <<<(NTILES + 3) / 4,    128, 0, stream>>>(h, W1, b1, W2, b2);
    k_stats  <<<STAT_BLOCKS,         256, 0, stream>>>(h, stats);
    k_final  <<<(NN * DD / 4) / 256, 256, 0, stream>>>(feat, h, gamma, beta, stats, out);
}